// GraphAttention_61409442398974
// MI455X (gfx1250) — compile-verified
//
#include <hip/hip_runtime.h>

// GraphAttention fused kernel for MI455X (gfx1250, wave32, WMMA).
//
// Algebraic refactor: scores = feats·(Q@Wk^T) and ctx = (Σ w_m f_m)@Wv + bv,
// which removes the 18 GFLOP K/V projections and makes the kernel purely
// memory-bound (~150 MB total -> ~6.4 us at 23.3 TB/s). Remaining GEMMs run
// on V_WMMA_F32_16X16X4_F32 (fp32 in/out, full precision).
// bk cancels (uniform shift under softmax), so it is intentionally unused.
//
// fbar is built by a transposed, coalesced re-read of the (WGP$-resident)
// neighbor tile with v_readlane broadcast of the softmax weights — far
// cheaper than a 68-wide cross-lane butterfly and keeps VGPR pressure low.

typedef float v2f __attribute__((ext_vector_type(2)));
typedef float v8f __attribute__((ext_vector_type(8)));

#define INTENT 64
#define HID 128
#define DIN 68           // INTENT + 4
#define MNEI 32
#define ROWS_PER_WAVE 16
#define WAVES_PER_BLOCK 4
#define ROWS_PER_BLOCK (ROWS_PER_WAVE * WAVES_PER_BLOCK)
#define PLD 80           // padded LDS stride for the 68-wide P rows

#define WMMA_F32(acc, a, b)                                                  \
  acc = __builtin_amdgcn_wmma_f32_16x16x4_f32(false, (a), false, (b),        \
                                              (short)0, (acc), false, false)

static __device__ __forceinline__ float readlane_f(float v, int l) {
  return __uint_as_float(__builtin_amdgcn_readlane(__float_as_uint(v), l));
}

__global__ __launch_bounds__(128) void ga_fused(
    const float* __restrict__ ego, const float* __restrict__ nz,
    const float* __restrict__ rp, const int* __restrict__ mask,
    const float* __restrict__ Wq, const float* __restrict__ bq,
    const float* __restrict__ Wk, const float* __restrict__ Wv,
    const float* __restrict__ bv, const float* __restrict__ Wo,
    const float* __restrict__ bo, float* __restrict__ out) {
  // Per-wave staging: bufQ (16x128, reused for fbar then ctx) + bufP (16x80).
  __shared__ float smem[WAVES_PER_BLOCK * (ROWS_PER_WAVE * HID + ROWS_PER_WAVE * PLD)];

  const int tid = threadIdx.x;
  const int lane = tid & 31;
  const int wv = tid >> 5;
  const int hf = lane >> 4;    // lane half: 0 -> K{0,1}, 1 -> K{2,3}
  const int l16 = lane & 15;
  const int kofs = 2 * hf;

  float* bufQ = smem + wv * (ROWS_PER_WAVE * HID + ROWS_PER_WAVE * PLD);
  float* bufP = bufQ + ROWS_PER_WAVE * HID;

  const int r0 = (blockIdx.x * WAVES_PER_BLOCK + wv) * ROWS_PER_WAVE;

  // ---------- Phase 1: Q = ego(16x64) @ Wq(64x128) + bq -> bufQ ----------
  {
    v8f accQ[8] = {};
    const float* Arow = ego + (size_t)(r0 + l16) * INTENT + kofs;
    for (int k0 = 0; k0 < INTENT; k0 += 4) {
      v2f a;
      a.x = Arow[k0];
      a.y = Arow[k0 + 1];
      const int kr = k0 + kofs;
#pragma unroll
      for (int t = 0; t < 8; ++t) {
        const int n = t * 16 + l16;
        v2f b;
        b.x = Wq[kr * HID + n];
        b.y = Wq[(kr + 1) * HID + n];
        WMMA_F32(accQ[t], a, b);
      }
    }
#pragma unroll
    for (int t = 0; t < 8; ++t) {
      const int n = t * 16 + l16;
      const float bias = bq[n];
#pragma unroll
      for (int j = 0; j < 8; ++j)
        bufQ[(j + 8 * hf) * HID + n] = accQ[t][j] + bias;
    }
  }
  __syncthreads();

  // ---------- Phase 2: P = Q(16x128) @ Wk^T(128x68) * rsqrt(H) -> bufP ----
  {
    v8f accP[5] = {};
    for (int k0 = 0; k0 < HID; k0 += 4) {
      v2f a;
      a.x = bufQ[l16 * HID + k0 + kofs];
      a.y = bufQ[l16 * HID + k0 + kofs + 1];
      const int kr = k0 + kofs;
#pragma unroll
      for (int t = 0; t < 5; ++t) {
        const int n = t * 16 + l16;
        v2f b;
        if (n < DIN) {           // Wk^T[k][n] = Wk[n][k]; zero-pad n>=68
          b.x = Wk[n * HID + kr];
          b.y = Wk[n * HID + kr + 1];
        } else {
          b.x = 0.f;
          b.y = 0.f;
        }
        WMMA_F32(accP[t], a, b);
      }
    }
    const float rscale = 0.08838834764831845f;  // 1/sqrt(128)
#pragma unroll
    for (int t = 0; t < 5; ++t) {
#pragma unroll
      for (int j = 0; j < 8; ++j)
        bufP[(j + 8 * hf) * PLD + t * 16 + l16] = accP[t][j] * rscale;
    }
  }
  __syncthreads();

  // ---------- Phase 3: scores -> softmax -> fbar ----------
  // lane m owns neighbor m during scoring; fbar uses a transposed coalesced
  // re-read (WGP$/L2 hit) with readlane-broadcast weights.
  for (int bl = 0; bl < ROWS_PER_WAVE; ++bl) {
    const size_t row = (size_t)(r0 + bl) * MNEI + lane;
    const float4* z4 = (const float4*)(nz + row * INTENT);
    const float4 r4 = *(const float4*)(rp + row * 4);

    float s = 0.f;
#pragma unroll
    for (int i = 0; i < 16; ++i) {
      const float4 v = z4[i];
      s = fmaf(v.x, bufP[bl * PLD + 4 * i + 0],
          fmaf(v.y, bufP[bl * PLD + 4 * i + 1],
          fmaf(v.z, bufP[bl * PLD + 4 * i + 2],
          fmaf(v.w, bufP[bl * PLD + 4 * i + 3], s))));
    }
    s = fmaf(r4.x, bufP[bl * PLD + 64],
        fmaf(r4.y, bufP[bl * PLD + 65],
        fmaf(r4.z, bufP[bl * PLD + 66],
        fmaf(r4.w, bufP[bl * PLD + 67], s))));
    if (mask[row] == 0) s = -1e9f;

    if (bl + 1 < ROWS_PER_WAVE)  // keep the HBM stream ahead
      __builtin_prefetch(nz + (row + MNEI) * INTENT, 0, 0);

    // softmax over the 32 neighbors (one per lane)
    float mx = s;
#pragma unroll
    for (int off = 16; off >= 1; off >>= 1)
      mx = fmaxf(mx, __shfl_xor(mx, off, 32));
    const float e = __expf(s - mx);
    float sum = e;
#pragma unroll
    for (int off = 16; off >= 1; off >>= 1) sum += __shfl_xor(sum, off, 32);
    const float w = e / sum;

    // fbar[d] = sum_m w_m * feats[m][d], d = lane and lane+32 (coalesced)
    const float* ztile = nz + (size_t)(r0 + bl) * MNEI * INTENT;
    float fb0 = 0.f, fb1 = 0.f;
#pragma unroll
    for (int m = 0; m < MNEI; ++m) {
      const float wm = readlane_f(w, m);
      fb0 = fmaf(wm, ztile[m * INTENT + lane], fb0);
      fb1 = fmaf(wm, ztile[m * INTENT + 32 + lane], fb1);
    }
    // rel_pos part (d = 64..67): tiny 4-value butterfly from registers
    float wx = w * r4.x, wy = w * r4.y, wz = w * r4.z, ww = w * r4.w;
#pragma unroll
    for (int off = 16; off >= 1; off >>= 1) {
      wx += __shfl_xor(wx, off, 32);
      wy += __shfl_xor(wy, off, 32);
      wz += __shfl_xor(wz, off, 32);
      ww += __shfl_xor(ww, off, 32);
    }
    // Q tile is dead; reuse bufQ rows for fbar
    bufQ[bl * HID + lane] = fb0;
    bufQ[bl * HID + 32 + lane] = fb1;
    const float relv = (lane == 0) ? wx : (lane == 1) ? wy : (lane == 2) ? wz : ww;
    if (lane < 4) bufQ[bl * HID + 64 + lane] = relv;
  }
  __syncthreads();

  // ---------- Phase 4: ctx = fbar(16x68) @ Wv(68x128) + bv -> bufQ --------
  {
    v8f accC[8] = {};
    for (int k0 = 0; k0 < DIN; k0 += 4) {  // 68 = 17 * 4, exact
      v2f a;
      a.x = bufQ[l16 * HID + k0 + kofs];
      a.y = bufQ[l16 * HID + k0 + kofs + 1];
      const int kr = k0 + kofs;
#pragma unroll
      for (int t = 0; t < 8; ++t) {
        const int n = t * 16 + l16;
        v2f b;
        b.x = Wv[kr * HID + n];
        b.y = Wv[(kr + 1) * HID + n];
        WMMA_F32(accC[t], a, b);
      }
    }
    __syncthreads();
#pragma unroll
    for (int t = 0; t < 8; ++t) {
      const int n = t * 16 + l16;
      const float bias = bv[n];
#pragma unroll
      for (int j = 0; j < 8; ++j)
        bufQ[(j + 8 * hf) * HID + n] = accC[t][j] + bias;
    }
  }
  __syncthreads();

  // ---------- Phase 5: out = ctx(16x128) @ Wo(128x64) + bo -> global ------
  {
    v8f accO[4] = {};
    for (int k0 = 0; k0 < HID; k0 += 4) {
      v2f a;
      a.x = bufQ[l16 * HID + k0 + kofs];
      a.y = bufQ[l16 * HID + k0 + kofs + 1];
      const int kr = k0 + kofs;
#pragma unroll
      for (int t = 0; t < 4; ++t) {
        const int n = t * 16 + l16;
        v2f b;
        b.x = Wo[kr * INTENT + n];
        b.y = Wo[(kr + 1) * INTENT + n];
        WMMA_F32(accO[t], a, b);
      }
    }
#pragma unroll
    for (int t = 0; t < 4; ++t) {
      const int n = t * 16 + l16;
      const float bias = bo[n];
#pragma unroll
      for (int j = 0; j < 8; ++j)
        out[(size_t)(r0 + j + 8 * hf) * INTENT + n] = accO[t][j] + bias;
    }
  }
}

extern "C" void kernel_launch(void* const* d_in, const int* in_sizes, int n_in,
                              void* d_out, int out_size, void* d_ws,
                              size_t ws_size, hipStream_t stream) {
  const float* ego = (const float*)d_in[0];
  const float* nz  = (const float*)d_in[1];
  const float* rp  = (const float*)d_in[2];
  const int* mask  = (const int*)d_in[3];
  const float* Wq  = (const float*)d_in[4];
  const float* bq  = (const float*)d_in[5];
  const float* Wk  = (const float*)d_in[6];
  // d_in[7] = bk: adds a per-row constant to all scores -> softmax-invariant.
  const float* Wv  = (const float*)d_in[8];
  const float* bv  = (const float*)d_in[9];
  const float* Wo  = (const float*)d_in[10];
  const float* bo  = (const float*)d_in[11];
  float* out = (float*)d_out;

  const int B = in_sizes[0] / INTENT;       // 16384
  const int blocks = B / ROWS_PER_BLOCK;    // 256 (B divisible by 64)
  ga_fused<<<blocks, WAVES_PER_BLOCK * 32, 0, stream>>>(
      ego, nz, rp, mask, Wq, bq, Wk, Wv, bv, Wo, bo, out);
}